// BasicLSTM_52355651338812
// MI455X (gfx1250) — compile-verified
//
#include <hip/hip_runtime.h>
#include <math.h>

// ---------------------------------------------------------------------------
// 2-layer LSTM (B=128,T=512,D=256,H=512) + FC head, MI455X / gfx1250.
//
// Design (see analysis): single persistent kernel, 128 WGs x 256 threads
// (8 wave32 waves).  Per step:
//   L0: g = [x_t ; h0(t-1)] @ [W_ih0 ; W_hh0]^T   (K=768, f32 WMMA 16x16x4)
//       -> gates -> c0,h0(t)          ... grid barrier ...
//   L1: g = [h0(t) ; h1(t-1)] @ [W_ih1 ; W_hh1]^T (K=1024)
//       -> gates -> c1,h1(t)          ... grid barrier ...
// Only final h0,h1 feed the FC head (reference discards ys).
// h double-buffered in workspace; A-panel staged in LDS (stride 1028 dwords,
// bank-conflict-free); weights stream from L2 (everything L2-resident).
// ---------------------------------------------------------------------------

typedef float v2f __attribute__((ext_vector_type(2)));
typedef float v8f __attribute__((ext_vector_type(8)));

#define D_IN   256
#define HID    512
#define BATCH  128
#define SEQ    512
#define NWG    128
#define NTHR   256
#define APAD   1028              // 16-row A panel stride in dwords (4-dword pad)

// workspace layout in floats
#define WS_H0  0                 // [2][128*512]
#define WS_H1  131072            // [2][128*512]
#define WS_C0  262144            // [128*512]
#define WS_C1  327680            // [128*512]
#define WS_CNT 393216            // barrier counter (1 dword)
#define WS_FLOATS 393217

__device__ __forceinline__ float sigmoidf_(float x) {
  return 1.0f / (1.0f + __expf(-x));
}

// one 16x16 gate tile: acc += A[16,K0+K1] x B[K0+K1,16] via f32 WMMA, K-step 4.
// Ap = A_lds + row*APAD + koff ; w0/w1 = weight row base + koff.
__device__ __forceinline__ v8f tile_gemm(const float* __restrict__ Ap,
                                         const float* __restrict__ w0, int K0,
                                         const float* __restrict__ w1, int K1,
                                         v8f acc) {
#pragma unroll 8
  for (int k = 0; k < K0; k += 4) {
    v2f a = *(const v2f*)(Ap + k);
    v2f b = *(const v2f*)(w0 + k);
    acc = __builtin_amdgcn_wmma_f32_16x16x4_f32(false, a, false, b,
                                                (short)0, acc, false, false);
  }
#pragma unroll 8
  for (int k = 0; k < K1; k += 4) {
    v2f a = *(const v2f*)(Ap + K0 + k);
    v2f b = *(const v2f*)(w1 + k);
    acc = __builtin_amdgcn_wmma_f32_16x16x4_f32(false, a, false, b,
                                                (short)0, acc, false, false);
  }
  return acc;
}

// stage 16 rows of [s0(l0 floats) ; s1(l1 floats)] into A_lds, float4 chunks.
__device__ __forceinline__ void stage_panel(float* __restrict__ A,
                                            const float* __restrict__ s0,
                                            size_t rs0, int l0,
                                            const float* __restrict__ s1,
                                            size_t rs1, int l1) {
  const int per = (l0 + l1) >> 2;          // float4s per row (literal at call)
  for (int i = threadIdx.x; i < 16 * per; i += NTHR) {
    int r = i / per;
    int c = (i - r * per) << 2;
    float4 v;
    if (c < l0) v = *(const float4*)(s0 + (size_t)r * rs0 + c);
    else        v = *(const float4*)(s1 + (size_t)r * rs1 + (c - l0));
    *(float4*)(A + (size_t)r * APAD + c) = v;
  }
}

// device-scope sense-free grid barrier (monotonic generation counter).
__device__ __forceinline__ void grid_barrier(unsigned* cnt, unsigned target) {
  __syncthreads();
  if (threadIdx.x == 0) {
    __hip_atomic_fetch_add(cnt, 1u, __ATOMIC_RELEASE, __HIP_MEMORY_SCOPE_AGENT);
    while (__hip_atomic_load(cnt, __ATOMIC_ACQUIRE,
                             __HIP_MEMORY_SCOPE_AGENT) < target)
      __builtin_amdgcn_s_sleep(2);
  }
  __syncthreads();
}

__global__ void init_ws(float* __restrict__ ws) {
  for (size_t i = (size_t)blockIdx.x * blockDim.x + threadIdx.x; i < WS_FLOATS;
       i += (size_t)gridDim.x * blockDim.x)
    ws[i] = 0.0f;
}

__global__ __launch_bounds__(NTHR, 1) void lstm_persistent(
    const float* __restrict__ x,
    const float* __restrict__ w_ih0, const float* __restrict__ w_hh0,
    const float* __restrict__ b0,
    const float* __restrict__ w_ih1, const float* __restrict__ w_hh1,
    const float* __restrict__ b1,
    const float* __restrict__ fc_w, const float* __restrict__ fc_b,
    float* __restrict__ out, float* __restrict__ ws) {
  __shared__ float A_lds[16 * APAD];   // 64.25 KB: A panel [16 x K<=1024]
  __shared__ float g_lds[8 * 256];     // 8 KB: 2 tiles x 4 gates x 16x16

  float*    h0  = ws + WS_H0;          // [2][B*H]
  float*    h1  = ws + WS_H1;
  float*    c0  = ws + WS_C0;
  float*    c1  = ws + WS_C1;
  unsigned* cnt = (unsigned*)(ws + WS_CNT);

  const int bt   = blockIdx.x & 7;     // batch tile (16 rows)
  const int cp   = blockIdx.x >> 3;    // column pair (32 hidden cols)
  const int lane = threadIdx.x & 31;
  const int wv   = threadIdx.x >> 5;   // wave 0..7
  const int htile = wv >> 2;           // 0..1
  const int gate  = wv & 3;            // i,f,g,o
  const int colbase = cp * 32 + htile * 16;
  const int nglob   = gate * HID + colbase + (lane & 15);  // W row (B col)
  const int arow    = lane & 15;                           // A row (M)
  const int koff    = (lane >> 4) << 1;                    // K sub-offset 0/2

  unsigned nbar = 0;

  for (int t = 0; t < SEQ; ++t) {
    const int wr = t & 1, rd = (t + 1) & 1;

    // ---------------- layer 0 ----------------
    {
      const float* xrow = x + ((size_t)bt * 16 * SEQ + t) * D_IN;
      const float* hrow = h0 + (size_t)rd * (BATCH * HID) +
                          (size_t)bt * 16 * HID;
      stage_panel(A_lds, xrow, (size_t)SEQ * D_IN, D_IN, hrow, HID, HID);
      __syncthreads();

      v8f acc = {};
      acc = tile_gemm(A_lds + (size_t)arow * APAD + koff,
                      w_ih0 + (size_t)nglob * D_IN + koff, D_IN,
                      w_hh0 + (size_t)nglob * HID + koff, HID, acc);

      union { v8f v; float f[8]; } u; u.v = acc;
      {
        float* gp = g_lds + wv * 256 + (lane & 15);
        const int mbase = (lane >> 4) * 8;
#pragma unroll
        for (int vv = 0; vv < 8; ++vv) gp[(mbase + vv) * 16] = u.f[vv];
      }
      __syncthreads();

      for (int e = threadIdx.x; e < 512; e += NTHR) {
        int tl = e >> 8, idx = e & 255;
        int m = idx >> 4, n = idx & 15;
        int j = cp * 32 + tl * 16 + n;
        int b = bt * 16 + m;
        float gi = g_lds[(tl * 4 + 0) * 256 + idx] + b0[j];
        float gf = g_lds[(tl * 4 + 1) * 256 + idx] + b0[HID + j];
        float gg = g_lds[(tl * 4 + 2) * 256 + idx] + b0[2 * HID + j];
        float go = g_lds[(tl * 4 + 3) * 256 + idx] + b0[3 * HID + j];
        size_t off = (size_t)b * HID + j;
        float cn = sigmoidf_(gf) * c0[off] + sigmoidf_(gi) * tanhf(gg);
        c0[off] = cn;
        h0[(size_t)wr * (BATCH * HID) + off] = sigmoidf_(go) * tanhf(cn);
      }
    }
    ++nbar; grid_barrier(cnt, nbar * NWG);   // h0(t) visible device-wide

    // ---------------- layer 1 ----------------
    {
      const float* h0row = h0 + (size_t)wr * (BATCH * HID) +
                           (size_t)bt * 16 * HID;
      const float* h1row = h1 + (size_t)rd * (BATCH * HID) +
                           (size_t)bt * 16 * HID;
      stage_panel(A_lds, h0row, HID, HID, h1row, HID, HID);
      __syncthreads();

      v8f acc = {};
      acc = tile_gemm(A_lds + (size_t)arow * APAD + koff,
                      w_ih1 + (size_t)nglob * HID + koff, HID,
                      w_hh1 + (size_t)nglob * HID + koff, HID, acc);

      union { v8f v; float f[8]; } u; u.v = acc;
      {
        float* gp = g_lds + wv * 256 + (lane & 15);
        const int mbase = (lane >> 4) * 8;
#pragma unroll
        for (int vv = 0; vv < 8; ++vv) gp[(mbase + vv) * 16] = u.f[vv];
      }
      __syncthreads();

      for (int e = threadIdx.x; e < 512; e += NTHR) {
        int tl = e >> 8, idx = e & 255;
        int m = idx >> 4, n = idx & 15;
        int j = cp * 32 + tl * 16 + n;
        int b = bt * 16 + m;
        float gi = g_lds[(tl * 4 + 0) * 256 + idx] + b1[j];
        float gf = g_lds[(tl * 4 + 1) * 256 + idx] + b1[HID + j];
        float gg = g_lds[(tl * 4 + 2) * 256 + idx] + b1[2 * HID + j];
        float go = g_lds[(tl * 4 + 3) * 256 + idx] + b1[3 * HID + j];
        size_t off = (size_t)b * HID + j;
        float cn = sigmoidf_(gf) * c1[off] + sigmoidf_(gi) * tanhf(gg);
        c1[off] = cn;
        h1[(size_t)wr * (BATCH * HID) + off] = sigmoidf_(go) * tanhf(cn);
      }
    }
    ++nbar; grid_barrier(cnt, nbar * NWG);   // h1(t) visible device-wide
  }

  // FC head: out[l*128+b] = h_final_l[b,:] . fc_w + fc_b.
  // final h lives in buffer index (SEQ-1)&1 == 1.
  if (blockIdx.x == 0) {
    const float fb = fc_b[0];
    for (int e = threadIdx.x; e < 2 * BATCH; e += NTHR) {
      int l = e >> 7, b = e & 127;
      const float* hp = (l ? h1 : h0) + (size_t)(BATCH * HID) +
                        (size_t)b * HID;
      float s = fb;
      for (int k = 0; k < HID; ++k) s = fmaf(hp[k], fc_w[k], s);
      out[e] = s;
    }
  }
}

extern "C" void kernel_launch(void* const* d_in, const int* in_sizes, int n_in,
                              void* d_out, int out_size, void* d_ws,
                              size_t ws_size, hipStream_t stream) {
  (void)in_sizes; (void)n_in; (void)out_size; (void)ws_size;
  const float* x     = (const float*)d_in[0];
  const float* w_ih0 = (const float*)d_in[1];
  const float* w_hh0 = (const float*)d_in[2];
  const float* b0    = (const float*)d_in[3];
  const float* w_ih1 = (const float*)d_in[4];
  const float* w_hh1 = (const float*)d_in[5];
  const float* b1    = (const float*)d_in[6];
  const float* fc_w  = (const float*)d_in[7];
  const float* fc_b  = (const float*)d_in[8];
  float* out = (float*)d_out;
  float* ws  = (float*)d_ws;

  init_ws<<<256, 256, 0, stream>>>(ws);   // zero h/c state + barrier counter
  lstm_persistent<<<NWG, NTHR, 0, stream>>>(x, w_ih0, w_hh0, b0, w_ih1, w_hh1,
                                            b1, fc_w, fc_b, out, ws);
}